// PreTrainGNN_7464653160644
// MI455X (gfx1250) — compile-verified
//
#include <hip/hip_runtime.h>
#include <hip/hip_bf16.h>

typedef __attribute__((ext_vector_type(2))) float v2f;
typedef __attribute__((ext_vector_type(8))) float v8f;

#define DCOLS 512
#define TM 128
#define TN 64
#define TK 64
#define AP (TK + 4)   // LDS row pitch: 68 floats -> 4-bank skew per row, float2/float4 aligned
#define NEG_SLOPE 0.01f

// ---------------------------------------------------------------------------
// GEMM: H[nrows x 512] = X[nrows x 512] @ W[512 x 512], fp32 WMMA 16x16x4
// Block: 256 threads (8 wave32), tile 128x64, K-block 64.
// ---------------------------------------------------------------------------
__global__ __launch_bounds__(256) void gcn_gemm(const float* __restrict__ X,
                                                const float* __restrict__ W,
                                                float* __restrict__ H,
                                                int nrows) {
  __shared__ float As[TM][AP];   // A tile, row-major
  __shared__ float Bs[TN][AP];   // W tile, transposed: Bs[n][k]

  const int tid  = threadIdx.x;
  const int wave = tid >> 5;
  const int lane = tid & 31;
  const int half = lane >> 4;    // 0: lanes 0-15, 1: lanes 16-31
  const int l16  = lane & 15;
  const int m0   = blockIdx.x * TM;
  const int n0   = blockIdx.y * TN;

  v8f acc0 = {}, acc1 = {}, acc2 = {}, acc3 = {};

  for (int kb = 0; kb < DCOLS; kb += TK) {
    // --- load A tile: 128x64 floats, 8 x float4 per thread, row-guarded ---
#pragma unroll
    for (int i = 0; i < 8; ++i) {
      int q  = tid + i * 256;
      int r  = q >> 4;            // 16 float4 per row
      int c4 = (q & 15) << 2;
      int rg = m0 + r;
      float4 v = make_float4(0.f, 0.f, 0.f, 0.f);
      if (rg < nrows) v = *(const float4*)(X + (size_t)rg * DCOLS + kb + c4);
      As[r][c4 + 0] = v.x; As[r][c4 + 1] = v.y;
      As[r][c4 + 2] = v.z; As[r][c4 + 3] = v.w;
    }
    // --- load B tile transposed: W[kb+k][n0+n] -> Bs[n][k], coalesced reads ---
#pragma unroll
    for (int i = 0; i < 16; ++i) {
      int q = tid + i * 256;
      int k = q >> 6;             // / 64
      int n = q & 63;
      Bs[n][k] = W[(size_t)(kb + k) * DCOLS + n0 + n];
    }
    __syncthreads();

    const int ar = wave * 16 + l16;   // A row this lane feeds
#pragma unroll
    for (int kk = 0; kk < TK; kk += 4) {
      // A frag: lanes 0-15 hold (K=kk,kk+1), lanes 16-31 hold (K=kk+2,kk+3)
      v2f a  = *(const v2f*)&As[ar][kk + 2 * half];
      // B frag (same half-wave K split), N = l16
      v2f b0 = *(const v2f*)&Bs[ 0 + l16][kk + 2 * half];
      v2f b1 = *(const v2f*)&Bs[16 + l16][kk + 2 * half];
      v2f b2 = *(const v2f*)&Bs[32 + l16][kk + 2 * half];
      v2f b3 = *(const v2f*)&Bs[48 + l16][kk + 2 * half];
      acc0 = __builtin_amdgcn_wmma_f32_16x16x4_f32(false, a, false, b0, (short)0, acc0, false, false);
      acc1 = __builtin_amdgcn_wmma_f32_16x16x4_f32(false, a, false, b1, (short)0, acc1, false, false);
      acc2 = __builtin_amdgcn_wmma_f32_16x16x4_f32(false, a, false, b2, (short)0, acc2, false, false);
      acc3 = __builtin_amdgcn_wmma_f32_16x16x4_f32(false, a, false, b3, (short)0, acc3, false, false);
    }
    __syncthreads();
  }

  // --- store: C/D layout: VGPR r = row (r + 8*half), N = l16 ---
#pragma unroll
  for (int r = 0; r < 8; ++r) {
    int rg = m0 + wave * 16 + r + 8 * half;
    if (rg < nrows) {
      float* o = H + (size_t)rg * DCOLS + n0 + l16;
      o[0]  = acc0[r];
      o[16] = acc1[r];
      o[32] = acc2[r];
      o[48] = acc3[r];
    }
  }
}

// ---------------------------------------------------------------------------
// Degree / normalization precompute
// ---------------------------------------------------------------------------
__global__ void init_deg(float* deg, int n) {
  int i = blockIdx.x * blockDim.x + threadIdx.x;
  if (i < n) deg[i] = 1.0f;   // self-loop contributes 1 to every node's degree
}

__global__ void deg_accum(const int* __restrict__ ei, int E, float* deg) {
  int e = blockIdx.x * blockDim.x + threadIdx.x;
  if (e < E) atomicAdd(&deg[ei[E + e]], 1.0f);   // col = targets
}

__global__ void deg_to_dinv(float* deg, int n) {
  int i = blockIdx.x * blockDim.x + threadIdx.x;
  if (i < n) deg[i] = rsqrtf(deg[i]);            // deg >= 1 always
}

__global__ void edge_norm(const int* __restrict__ ei, int E,
                          const float* __restrict__ dinv, float* __restrict__ nrm) {
  int e = blockIdx.x * blockDim.x + threadIdx.x;
  if (e < E) nrm[e] = dinv[ei[e]] * dinv[ei[E + e]];
}

// ---------------------------------------------------------------------------
// agg[i,:] = dinv[i]^2 * h[i,:]   (self-loop term; also fully initializes agg)
// ---------------------------------------------------------------------------
__global__ void self_loop_init(const float* __restrict__ h, const float* __restrict__ dinv,
                               float* __restrict__ agg, long total4) {
  long i = (long)blockIdx.x * blockDim.x + threadIdx.x;
  if (i >= total4) return;
  int row = (int)(i >> 7);                    // 512/4 = 128 float4 per row
  float s = dinv[row];
  s = s * s;
  float4 v = ((const float4*)h)[i];
  v.x *= s; v.y *= s; v.z *= s; v.w *= s;
  ((float4*)agg)[i] = v;
}

// ---------------------------------------------------------------------------
// Edge scatter: one wave32 per edge, agg[col,:] += norm[e] * h[row,:]
// L2-resident float atomics, coalesced 128B stripes per wave.
// ---------------------------------------------------------------------------
__global__ void edge_scatter(const int* __restrict__ ei, int E,
                             const float* __restrict__ nrm,
                             const float* __restrict__ h,
                             float* __restrict__ agg) {
  int gtid = blockIdx.x * blockDim.x + threadIdx.x;
  int wid  = gtid >> 5;
  int lane = threadIdx.x & 31;
  int nw   = (gridDim.x * blockDim.x) >> 5;
  for (int e = wid; e < E; e += nw) {
    int src = ei[e];
    int dst = ei[E + e];
    float s = nrm[e];
    const float* hs = h + (size_t)src * DCOLS;
    float* ad = agg + (size_t)dst * DCOLS;
#pragma unroll 4
    for (int d = lane; d < DCOLS; d += 32)
      atomicAdd(&ad[d], s * hs[d]);
  }
}

// ---------------------------------------------------------------------------
// out = leaky_relu(agg + bias)
// ---------------------------------------------------------------------------
__global__ void bias_lrelu(const float* __restrict__ agg, const float* __restrict__ bias,
                           float* __restrict__ out, long total4) {
  long i = (long)blockIdx.x * blockDim.x + threadIdx.x;
  if (i >= total4) return;
  int c = (int)((i << 2) & (DCOLS - 1));
  float4 v = ((const float4*)agg)[i];
  float4 b = *(const float4*)(bias + c);
  v.x += b.x; v.y += b.y; v.z += b.z; v.w += b.w;
  v.x = v.x > 0.f ? v.x : NEG_SLOPE * v.x;
  v.y = v.y > 0.f ? v.y : NEG_SLOPE * v.y;
  v.z = v.z > 0.f ? v.z : NEG_SLOPE * v.z;
  v.w = v.w > 0.f ? v.w : NEG_SLOPE * v.w;
  ((float4*)out)[i] = v;
}

// ---------------------------------------------------------------------------
extern "C" void kernel_launch(void* const* d_in, const int* in_sizes, int n_in,
                              void* d_out, int out_size, void* d_ws, size_t ws_size,
                              hipStream_t stream) {
  const float* x   = (const float*)d_in[0];
  const int*   ei  = (const int*)d_in[1];      // [2, E] flat: rows then cols
  const float* Ws  = (const float*)d_in[2];    // [L, 512, 512]
  const float* bs  = (const float*)d_in[3];    // [L, 512]

  const int N = in_sizes[0] / DCOLS;
  const int E = in_sizes[1] / 2;
  const int L = in_sizes[2] / (DCOLS * DCOLS);

  // workspace: A (x ping-pong) | B (h) | dinv | norm
  float* A    = (float*)d_ws;
  float* B    = A + (size_t)N * DCOLS;
  float* dinv = B + (size_t)N * DCOLS;
  float* nrm  = dinv + N;

  const long total4 = (long)N * DCOLS / 4;
  const int  tpb = 256;

  // --- precompute symmetric normalization (deg includes self-loops) ---
  init_deg<<<(N + tpb - 1) / tpb, tpb, 0, stream>>>(dinv, N);
  deg_accum<<<(E + tpb - 1) / tpb, tpb, 0, stream>>>(ei, E, dinv);
  deg_to_dinv<<<(N + tpb - 1) / tpb, tpb, 0, stream>>>(dinv, N);
  edge_norm<<<(E + tpb - 1) / tpb, tpb, 0, stream>>>(ei, E, dinv, nrm);

  dim3 ggrid((N + TM - 1) / TM, DCOLS / TN);
  const int sgrid = 2048;                                  // 16384 waves grid-stride
  const int egrid = (int)((total4 + tpb - 1) / tpb);

  const float* xin = x;
  for (int l = 0; l < L; ++l) {
    gcn_gemm<<<ggrid, tpb, 0, stream>>>(xin, Ws + (size_t)l * DCOLS * DCOLS, B, N);
    self_loop_init<<<egrid, tpb, 0, stream>>>(B, dinv, A, total4);
    edge_scatter<<<sgrid, tpb, 0, stream>>>(ei, E, nrm, B, A);
    float* xout = (l == L - 1) ? (float*)d_out : A;        // in-place relu on A is safe
    bias_lrelu<<<egrid, tpb, 0, stream>>>(A, bs + (size_t)l * DCOLS, xout, total4);
    xin = A;
  }
}